// CROM_Layer_81140522156285
// MI455X (gfx1250) — compile-verified
//
#include <hip/hip_runtime.h>
#include <hip/hip_bf16.h>
#include <math.h>

#define DIM 1024
#define SEQ 8192
#define BATCH 4

typedef __attribute__((ext_vector_type(2))) float v2f;
typedef __attribute__((ext_vector_type(8))) float v8f;

// ---------------------------------------------------------------------------
// prep: zero the padded 16xD A-tiles, fill rows 0..3 of last16 with x[:, -1, :]
// ---------------------------------------------------------------------------
__global__ __launch_bounds__(256) void prep_kernel(const float* __restrict__ x,
                                                   float* __restrict__ last16,
                                                   float* __restrict__ c16) {
    int idx = blockIdx.x * 256 + threadIdx.x;   // 0 .. 16383
    float v = 0.0f;
    if (idx < BATCH * DIM) {
        int b = idx >> 10, d = idx & (DIM - 1);
        v = x[((size_t)b * SEQ + (SEQ - 1)) * DIM + d];
    }
    last16[idx] = v;
    c16[idx]    = 0.0f;
}

// ---------------------------------------------------------------------------
// 16xD @ (DxD) GEMM on the WMMA path: Out(16x1024) = A(16x1024) x B(1024x1024)
// TRANSB=true  -> B[k][n] = W[n][k]   (Out = A @ W^T)
// TRANSB=false -> B[k][n] = W[k][n]   (Out = A @ W)
// One wave per 16-column tile; K-loop over DIM in steps of 4 using
// V_WMMA_F32_16X16X4_F32 (f32 in, f32 accumulate: exact).
// eid (device scalar) optionally offsets W / bias for the expert matmul.
// ---------------------------------------------------------------------------
template <bool TRANSB, bool BIAS>
__global__ __launch_bounds__(32) void gemm16_kernel(const float* __restrict__ A,
                                                    const float* __restrict__ W,
                                                    const float* __restrict__ bias,
                                                    const int*   __restrict__ eid,
                                                    float* __restrict__ O) {
    if (eid) {                       // uniform branch: EXEC stays all-ones
        size_t e = (size_t)(*eid);
        W += e * (size_t)DIM * DIM;
        if (BIAS) bias += e * (size_t)DIM;
    }
    const int ntile = blockIdx.x;            // 0..63
    const int lane  = threadIdx.x;           // 0..31 (wave32)
    const int mrow  = lane & 15;             // A: lane -> M row
    const int khalf = (lane >> 4) * 2;       // lanes 16..31 hold K=2,3
    const int ncol  = ntile * 16 + (lane & 15);  // B/C/D: lane -> N column

    v8f c = {};
    for (int k0 = 0; k0 < DIM; k0 += 4) {
        v2f a, b;
        // A 16x4 fragment: VGPR0 = A[m][k0+khalf], VGPR1 = A[m][k0+khalf+1]
        a.x = A[mrow * DIM + k0 + khalf];
        a.y = A[mrow * DIM + k0 + khalf + 1];
        if (TRANSB) {
            b.x = W[(size_t)ncol * DIM + k0 + khalf];
            b.y = W[(size_t)ncol * DIM + k0 + khalf + 1];
        } else {
            b.x = W[(size_t)(k0 + khalf) * DIM + ncol];
            b.y = W[(size_t)(k0 + khalf + 1) * DIM + ncol];
        }
        c = __builtin_amdgcn_wmma_f32_16x16x4_f32(
                /*neg_a=*/false, a, /*neg_b=*/false, b,
                /*c_mod=*/(short)0, c, /*reuse_a=*/false, /*reuse_b=*/false);
    }
    // C/D layout: VGPR i -> M = i + 8*(lane>=16), N = ncol
    const float bv   = BIAS ? bias[ncol] : 0.0f;
    const int   mbase = 8 * (lane >> 4);
#pragma unroll
    for (int i = 0; i < 8; ++i)
        O[(mbase + i) * DIM + ncol] = c[i] + bv;
}

// ---------------------------------------------------------------------------
// Fused pass 1 over x: copy x -> out (bulk of the output) and compute
// scores[b,s] = dot(x[b,s,:], r[b,:]) / sqrt(D).  One wave per row.
// ---------------------------------------------------------------------------
__global__ __launch_bounds__(256) void scores_copy_kernel(const float* __restrict__ x,
                                                          const float* __restrict__ r16,
                                                          float* __restrict__ out,
                                                          float* __restrict__ scores) {
    const int wave = blockIdx.x * (256 / 32) + (threadIdx.x >> 5);  // 0..32767 == b*SEQ+s
    const int lane = threadIdx.x & 31;
    const int b    = wave >> 13;

    const float4* __restrict__ xr = (const float4*)(x   + (size_t)wave * DIM);
    float4*       __restrict__ od = (float4*)      (out + (size_t)wave * DIM);
    const float4* __restrict__ rr = (const float4*)(r16 + b * DIM);

    float acc = 0.0f;
#pragma unroll
    for (int i = 0; i < 8; ++i) {
        float4 xv = xr[lane + i * 32];
        float4 rv = rr[lane + i * 32];
        od[lane + i * 32] = xv;
        acc += xv.x * rv.x + xv.y * rv.y + xv.z * rv.z + xv.w * rv.w;
    }
#pragma unroll
    for (int off = 16; off > 0; off >>= 1)
        acc += __shfl_down(acc, off, 32);
    if (lane == 0)
        scores[wave] = acc * (1.0f / 32.0f);   // 1/sqrt(1024)
}

// ---------------------------------------------------------------------------
// Softmax stats per batch: row max + sum(exp(s - max)). One block per batch.
// ---------------------------------------------------------------------------
__global__ __launch_bounds__(256) void softmax_stats_kernel(const float* __restrict__ scores,
                                                            float* __restrict__ stats) {
    const int b = blockIdx.x;
    const int t = threadIdx.x;
    __shared__ float red[256];
    const float* sc = scores + (size_t)b * SEQ;

    float m = -INFINITY;
    for (int i = t; i < SEQ; i += 256) m = fmaxf(m, sc[i]);
    red[t] = m; __syncthreads();
    for (int s2 = 128; s2 > 0; s2 >>= 1) {
        if (t < s2) red[t] = fmaxf(red[t], red[t + s2]);
        __syncthreads();
    }
    m = red[0]; __syncthreads();

    float sum = 0.0f;
    for (int i = t; i < SEQ; i += 256) sum += expf(sc[i] - m);
    red[t] = sum; __syncthreads();
    for (int s2 = 128; s2 > 0; s2 >>= 1) {
        if (t < s2) red[t] += red[t + s2];
        __syncthreads();
    }
    if (t == 0) { stats[b * 2] = m; stats[b * 2 + 1] = red[0]; }
}

// ---------------------------------------------------------------------------
// Pass 2 over x: partial attn-weighted sums. Grid = (b, s-chunk of 256 rows),
// each thread owns 4 contiguous columns (float4, fully coalesced).
// Deterministic: fixed-order accumulation + separate tree reduce (no FP atomics).
// ---------------------------------------------------------------------------
__global__ __launch_bounds__(256) void weighted_partial_kernel(const float* __restrict__ x,
                                                               const float* __restrict__ scores,
                                                               const float* __restrict__ stats,
                                                               float* __restrict__ partial) {
    const int b     = blockIdx.x >> 5;     // /32
    const int chunk = blockIdx.x & 31;
    const int t     = threadIdx.x;
    const float m    = stats[b * 2];
    const float invz = 1.0f / stats[b * 2 + 1];

    const float* __restrict__ xb = x + (size_t)b * SEQ * DIM;
    const int s0 = chunk * (SEQ / 32);
    float4 acc = {0.0f, 0.0f, 0.0f, 0.0f};
    for (int s = s0; s < s0 + (SEQ / 32); ++s) {
        __builtin_prefetch(xb + (size_t)(s + 4) * DIM + t * 4, 0, 1);
        const float w   = expf(scores[(size_t)b * SEQ + s] - m) * invz;
        const float4 xv = ((const float4*)(xb + (size_t)s * DIM))[t];
        acc.x += w * xv.x; acc.y += w * xv.y; acc.z += w * xv.z; acc.w += w * xv.w;
    }
    ((float4*)(partial + ((size_t)(b * 32 + chunk)) * DIM))[t] = acc;
}

__global__ __launch_bounds__(256) void reduce_partials_kernel(const float* __restrict__ partial,
                                                              float* __restrict__ c16) {
    const int idx = blockIdx.x * 256 + threadIdx.x;     // 0..4095
    const int b = idx >> 10, d = idx & (DIM - 1);
    float s = 0.0f;
    for (int j = 0; j < 32; ++j)
        s += partial[((size_t)(b * 32 + j)) * DIM + d];
    c16[b * DIM + d] = s;                               // rows 4..15 stay zero from prep
}

// ---------------------------------------------------------------------------
// Finalize: resid = last + expert_out, LayerNorm, write out[:, -1, :].
// ---------------------------------------------------------------------------
__global__ __launch_bounds__(256) void finalize_kernel(const float* __restrict__ last16,
                                                       const float* __restrict__ out16,
                                                       const float* __restrict__ gamma,
                                                       const float* __restrict__ beta,
                                                       float* __restrict__ out) {
    const int b = blockIdx.x;
    const int t = threadIdx.x;
    __shared__ float red[256];

    float r[4];
    float lsum = 0.0f;
#pragma unroll
    for (int j = 0; j < 4; ++j) {
        int d = t * 4 + j;
        r[j] = last16[b * DIM + d] + out16[b * DIM + d];
        lsum += r[j];
    }
    red[t] = lsum; __syncthreads();
    for (int s2 = 128; s2 > 0; s2 >>= 1) { if (t < s2) red[t] += red[t + s2]; __syncthreads(); }
    const float mu = red[0] * (1.0f / DIM); __syncthreads();

    float vs = 0.0f;
#pragma unroll
    for (int j = 0; j < 4; ++j) { float dd = r[j] - mu; vs += dd * dd; }
    red[t] = vs; __syncthreads();
    for (int s2 = 128; s2 > 0; s2 >>= 1) { if (t < s2) red[t] += red[t + s2]; __syncthreads(); }
    const float inv = rsqrtf(red[0] * (1.0f / DIM) + 1e-5f);

    float* o = out + ((size_t)b * SEQ + (SEQ - 1)) * DIM;
#pragma unroll
    for (int j = 0; j < 4; ++j) {
        int d = t * 4 + j;
        o[d] = (r[j] - mu) * inv * gamma[d] + beta[d];
    }
}

// ---------------------------------------------------------------------------
extern "C" void kernel_launch(void* const* d_in, const int* in_sizes, int n_in,
                              void* d_out, int out_size, void* d_ws, size_t ws_size,
                              hipStream_t stream) {
    (void)in_sizes; (void)n_in; (void)out_size; (void)ws_size;

    const float* x        = (const float*)d_in[0];
    const float* Wq       = (const float*)d_in[1];
    const float* Wk       = (const float*)d_in[2];
    const float* Wv       = (const float*)d_in[3];
    const float* expert_W = (const float*)d_in[4];
    const float* expert_b = (const float*)d_in[5];
    const float* ln_gamma = (const float*)d_in[6];
    const float* ln_beta  = (const float*)d_in[7];
    const int*   eid      = (const int*)  d_in[8];
    float* out = (float*)d_out;

    // workspace layout (floats)
    float* ws      = (float*)d_ws;
    float* last16  = ws;                    // 16*1024
    float* q16     = ws + 16384;            // 16*1024
    float* r16     = ws + 32768;            // 16*1024
    float* c16     = ws + 49152;            // 16*1024
    float* ctx16   = ws + 65536;            // 16*1024
    float* out16   = ws + 81920;            // 16*1024
    float* scores  = ws + 98304;            // 4*8192
    float* partial = ws + 131072;           // 4*32*1024
    float* stats   = ws + 262144;           // 4*2

    // 1) gather last tokens into padded 16xD tile, zero c16 pad rows
    prep_kernel<<<64, 256, 0, stream>>>(x, last16, c16);
    // 2) q = last @ Wq^T          (WMMA f32 16x16x4)
    gemm16_kernel<true,  false><<<64, 32, 0, stream>>>(last16, Wq, nullptr, nullptr, q16);
    // 3) r = q @ Wk               (folds k out of existence)
    gemm16_kernel<false, false><<<64, 32, 0, stream>>>(q16, Wk, nullptr, nullptr, r16);
    // 4) stream x once: copy to out + scores = (x . r)/sqrt(D)
    scores_copy_kernel<<<(BATCH * SEQ) / 8, 256, 0, stream>>>(x, r16, out, scores);
    // 5) softmax stats per batch
    softmax_stats_kernel<<<BATCH, 256, 0, stream>>>(scores, stats);
    // 6) stream x again: attn-weighted row sum (partials, deterministic)
    weighted_partial_kernel<<<BATCH * 32, 256, 0, stream>>>(x, scores, stats, partial);
    reduce_partials_kernel<<<16, 256, 0, stream>>>(partial, c16);
    // 7) context = c @ Wv^T       (folds v out of existence)
    gemm16_kernel<true,  false><<<64, 32, 0, stream>>>(c16, Wv, nullptr, nullptr, ctx16);
    // 8) expert: out = ctx @ W_e^T + b_e   (expert_id read on-device)
    gemm16_kernel<true,  true ><<<64, 32, 0, stream>>>(ctx16, expert_W, expert_b, eid, out16);
    // 9) residual + LayerNorm into out[:, -1, :]
    finalize_kernel<<<BATCH, 256, 0, stream>>>(last16, out16, ln_gamma, ln_beta, out);
}